// DeployModel_72267119723328
// MI455X (gfx1250) — compile-verified
//
#include <hip/hip_runtime.h>
#include <math.h>

// ---------------- problem constants (from reference) ----------------
#define B_        8
#define C_        1203
#define NANCH     8400      // 80*80 + 40*40 + 20*20
#define PRE_TOPK  1000
#define KEEP_TOPK 100
#define NSORT     16384     // next pow2 >= 8400
#define NPAD      1008      // 63 * 16 (tile-padded PRE_TOPK)
#define TILES     63
#define SCORE_THR 0.25f
#define IOU_THR_  0.65f
#define CLASS_OFF 4096.0f
#define NCHUNK    4
#define CPER      301       // ceil(1203/4)

typedef __attribute__((ext_vector_type(2))) float v2f;
typedef __attribute__((ext_vector_type(8))) float v8f;

// map flat anchor index -> (level HW, in-level index, width, stride, level id)
__device__ __forceinline__ void levelOf(int n, int& hw, int& s, int& w, float& strd, int& lvl) {
  if (n < 6400)      { hw = 6400; s = n;        w = 80; strd =  8.f; lvl = 0; }
  else if (n < 8000) { hw = 1600; s = n - 6400; w = 40; strd = 16.f; lvl = 1; }
  else               { hw =  400; s = n - 8000; w = 20; strd = 32.f; lvl = 2; }
}

// ---------------- stage 1a: partial max/argmax over a class chunk ----------------
// grid: (ceil(8400/256), B, NCHUNK); lanes map to contiguous spatial s -> fully
// coalesced 128B/wave loads; classes strided by HW. Streaming, memory bound.
__global__ void k_partial_max(const float* __restrict__ c0, const float* __restrict__ c1,
                              const float* __restrict__ c2,
                              float* __restrict__ pScore, int* __restrict__ pLabel) {
  int n = blockIdx.x * 256 + threadIdx.x;
  if (n >= NANCH) return;
  int b = blockIdx.y, chunk = blockIdx.z;
  int hw, s, w, lvl; float strd;
  levelOf(n, hw, s, w, strd, lvl);
  const float* base = (lvl == 0) ? c0 : (lvl == 1 ? c1 : c2);
  int cBeg = chunk * CPER;
  int cEnd = cBeg + CPER; if (cEnd > C_) cEnd = C_;
  const float* p = base + ((size_t)b * C_ + cBeg) * (size_t)hw + s;
  size_t st = (size_t)hw;
  float m = -3.4e38f; int arg = cBeg;
  int c = cBeg;
  for (; c + 4 <= cEnd; c += 4) {
    __builtin_prefetch(p + 16 * st, 0, 0);        // global_prefetch_b8, warm L2 ahead of stream
    float v0 = p[0], v1 = p[st], v2 = p[2 * st], v3 = p[3 * st];
    if (v0 > m) { m = v0; arg = c;     }
    if (v1 > m) { m = v1; arg = c + 1; }
    if (v2 > m) { m = v2; arg = c + 2; }
    if (v3 > m) { m = v3; arg = c + 3; }
    p += 4 * st;
  }
  for (; c < cEnd; ++c) { float v = p[0]; if (v > m) { m = v; arg = c; } p += st; }
  size_t o = ((size_t)chunk * B_ + b) * NANCH + n;
  pScore[o] = m; pLabel[o] = arg;
}

// ---------------- stage 1b: combine chunks, sigmoid(max), decode boxes ----------------
__global__ void k_reduce_decode(const float* __restrict__ bb0, const float* __restrict__ bb1,
                                const float* __restrict__ bb2,
                                const float* __restrict__ pScore, const int* __restrict__ pLabel,
                                float* __restrict__ scores, int* __restrict__ labels,
                                float* __restrict__ boxes) {
  int n = blockIdx.x * 256 + threadIdx.x;
  if (n >= NANCH) return;
  int b = blockIdx.y;
  float m = -3.4e38f; int lab = 0;
  for (int ch = 0; ch < NCHUNK; ++ch) {       // ascending chunk order + strict '>' keeps first-max tie rule
    size_t o = ((size_t)ch * B_ + b) * NANCH + n;
    float v = pScore[o];
    if (v > m) { m = v; lab = pLabel[o]; }
  }
  float score = 1.0f / (1.0f + expf(-m));     // sigmoid applied once per anchor (monotone)
  int hw, s, w, lvl; float strd;
  levelOf(n, hw, s, w, strd, lvl);
  const float* bb = (lvl == 0) ? bb0 : (lvl == 1 ? bb1 : bb2);
  size_t bo = (size_t)b * 4 * hw + s;
  float d0 = bb[bo]           * strd;
  float d1 = bb[bo +     hw]  * strd;
  float d2 = bb[bo + 2 * hw]  * strd;
  float d3 = bb[bo + 3 * hw]  * strd;
  int yy = s / w, xx = s - yy * w;
  float px = (xx + 0.5f) * strd, py = (yy + 0.5f) * strd;
  size_t o = (size_t)b * NANCH + n;
  scores[o] = score; labels[o] = lab;
  float* bx = boxes + o * 4;
  bx[0] = px - d0; bx[1] = py - d1; bx[2] = px + d2; bx[3] = py + d3;
}

// ---------------- stage 2: per-batch full bitonic sort in 128KB LDS ----------------
// key = (float-bits(score) << 32) | idx ; all scores are positive sigmoids so raw
// bits order correctly; pad = 0 sorts last. 320KB LDS/WGP makes 16K x 8B feasible.
__global__ void __launch_bounds__(1024) k_sort(const float* __restrict__ scores,
                                               int* __restrict__ topIdx,
                                               float* __restrict__ topScore) {
  extern __shared__ unsigned long long keys[];
  int b = blockIdx.x;
  for (unsigned k = threadIdx.x; k < NSORT; k += blockDim.x) {
    unsigned long long key = 0ull;
    if (k < NANCH) {
      unsigned sb = __float_as_uint(scores[(size_t)b * NANCH + k]);
      key = ((unsigned long long)sb << 32) | k;
    }
    keys[k] = key;
  }
  for (unsigned ksz = 2; ksz <= NSORT; ksz <<= 1) {
    for (unsigned j = ksz >> 1; j > 0; j >>= 1) {
      __syncthreads();
      for (unsigned i = threadIdx.x; i < NSORT; i += blockDim.x) {
        unsigned ixj = i ^ j;
        if (ixj > i) {
          unsigned long long a = keys[i], c = keys[ixj];
          bool top = ((i & ksz) == 0);          // descending overall
          if (top ? (a < c) : (a > c)) { keys[i] = c; keys[ixj] = a; }
        }
      }
    }
  }
  __syncthreads();
  for (unsigned r = threadIdx.x; r < PRE_TOPK; r += blockDim.x) {
    unsigned long long key = keys[r];
    topIdx[b * PRE_TOPK + r]   = (int)(unsigned)(key & 0xffffffffull);
    topScore[b * PRE_TOPK + r] = __uint_as_float((unsigned)(key >> 32));
  }
}

// ---------------- stage 3: NMS ----------------
// Phase B builds the full 1000x1000 suppression bitmask in LDS; the pairwise
// area-sum tile (union denominator precursor) is one V_WMMA_F32_16X16X4_F32
// outer product per 16x16 tile: A=[area_j,1,0,0], B=[1;area_i;0;0] -> D=area_j+area_i.
// Phase C runs the serial keep-scan entirely inside one wave32 on the bitmask.
__global__ void __launch_bounds__(1024) k_nms(const int* __restrict__ topIdx,
                                              const float* __restrict__ topScore,
                                              const float* __restrict__ boxes,
                                              const int* __restrict__ labels,
                                              float* __restrict__ out) {
  extern __shared__ char nmsMem[];
  float*    offs  = (float*)nmsMem;            // NPAD*4 : class-offset boxes
  float*    area  = offs + NPAD * 4;           // NPAD
  float*    tsc   = area + NPAD;               // NPAD   : sorted scores
  unsigned* sup   = (unsigned*)(tsc + NPAD);   // NPAD*32: suppression bitmask columns
  unsigned* keepw = sup + NPAD * 32;           // 32

  int b = blockIdx.x, tid = threadIdx.x;
  int lane = tid & 31;

  // Phase A: stage candidates in LDS
  if (tid < NPAD) {
    if (tid < PRE_TOPK) {
      int idx = topIdx[b * PRE_TOPK + tid];
      float sc = topScore[b * PRE_TOPK + tid];
      const float* bx = boxes + ((size_t)b * NANCH + idx) * 4;
      float off = (float)labels[(size_t)b * NANCH + idx] * CLASS_OFF; // class-aware trick
      float x1 = bx[0] + off, y1 = bx[1] + off, x2 = bx[2] + off, y2 = bx[3] + off;
      offs[tid * 4 + 0] = x1; offs[tid * 4 + 1] = y1;
      offs[tid * 4 + 2] = x2; offs[tid * 4 + 3] = y2;
      area[tid] = (x2 - x1) * (y2 - y1);
      tsc[tid]  = sc;
    } else {                                    // pad: zero-area far-away points -> iou == 0
      float p = -1.0e8f - (float)tid * 8192.0f;
      offs[tid * 4 + 0] = p; offs[tid * 4 + 1] = p;
      offs[tid * 4 + 2] = p; offs[tid * 4 + 3] = p;
      area[tid] = 0.f; tsc[tid] = -1.f;
    }
  }
  for (int w = tid; w < NPAD * 32; w += blockDim.x) sup[w] = 0u;
  __syncthreads();

  // Phase B: 63x63 tiles of 16x16 pairs; one wave per tile iteration (EXEC all ones)
  int wave = tid >> 5;
  int nWaves = blockDim.x >> 5;
  for (int t = wave; t < TILES * TILES; t += nWaves) {
    int tj = t / TILES, ti = t - tj * TILES;
    int J0 = tj * 16, I0 = ti * 16;
    v2f av, bv;
    if (lane < 16) { av.x = area[J0 + lane]; av.y = 1.0f;            // A: K=0,1
                     bv.x = 1.0f;            bv.y = area[I0 + lane]; } // B: K=0,1
    else           { av.x = 0.f; av.y = 0.f; bv.x = 0.f; bv.y = 0.f; } // K=2,3 zero
    v8f acc = {};
    acc = __builtin_amdgcn_wmma_f32_16x16x4_f32(
        /*neg_a=*/false, av, /*neg_b=*/false, bv,
        /*c_mod=*/(short)0, acc, /*reuse_a=*/false, /*reuse_b=*/false);
    // D layout: lanes 0-15 -> N=lane, M=vgpr; lanes 16-31 -> N=lane-16, M=8+vgpr
    int ii    = I0 + (lane & 15);
    int mbase = (lane < 16) ? 0 : 8;
    float ix1 = offs[ii * 4 + 0], iy1 = offs[ii * 4 + 1];
    float ix2 = offs[ii * 4 + 2], iy2 = offs[ii * 4 + 3];
    unsigned bits = 0u;
#pragma unroll
    for (int v = 0; v < 8; ++v) {
      int jj = J0 + mbase + v;
      float jx1 = offs[jj * 4 + 0], jy1 = offs[jj * 4 + 1];
      float jx2 = offs[jj * 4 + 2], jy2 = offs[jj * 4 + 3];
      float ww = fmaxf(fminf(jx2, ix2) - fmaxf(jx1, ix1), 0.f);
      float hh = fmaxf(fminf(jy2, iy2) - fmaxf(jy1, iy1), 0.f);
      float inter = ww * hh;
      float uni   = acc[v] - inter + 1e-7f;     // (area_j + area_i) from WMMA
      if (inter > IOU_THR_ * uni) bits |= (1u << v);   // iou > thr, division-free
    }
    if (bits) atomicOr(&sup[ii * 32 + (J0 >> 5)], bits << ((J0 & 31) + mbase));
  }
  __syncthreads();

  // Phase C: serial keep-scan, one wave32, keep bitmask one word per lane
  if (tid < 32) {
    unsigned kw = 0u;
    for (int i = 0; i < PRE_TOPK; ++i) {
      unsigned hit = sup[i * 32 + lane] & kw;
      bool suppressed = (__ballot(hit != 0u) != 0ull);
      bool keepI = (tsc[i] > SCORE_THR) && !suppressed;
      if (keepI && lane == (i >> 5)) kw |= (1u << (i & 31));
    }
    keepw[lane] = kw;
  }
  __syncthreads();

  // Phase D: compact (already score-sorted -> first 100 kept are the answer)
  if (tid == 0) {
    float* oNum    = out;
    float* oBoxes  = out + B_;
    float* oScores = out + B_ + B_ * KEEP_TOPK * 4;
    float* oLabels = out + B_ + B_ * KEEP_TOPK * 4 + B_ * KEEP_TOPK;
    int rank = 0;
    for (int k = 0; k < PRE_TOPK && rank < KEEP_TOPK; ++k) {
      if ((keepw[k >> 5] >> (k & 31)) & 1u) {
        int idx = topIdx[b * PRE_TOPK + k];
        const float* bx = boxes + ((size_t)b * NANCH + idx) * 4;
        float* ob = oBoxes + ((size_t)b * KEEP_TOPK + rank) * 4;
        ob[0] = bx[0]; ob[1] = bx[1]; ob[2] = bx[2]; ob[3] = bx[3];
        oScores[b * KEEP_TOPK + rank] = topScore[b * PRE_TOPK + k];
        oLabels[b * KEEP_TOPK + rank] = (float)labels[(size_t)b * NANCH + idx];
        rank++;
      }
    }
    oNum[b] = (float)rank;
    for (int r = rank; r < KEEP_TOPK; ++r) {
      float* ob = oBoxes + ((size_t)b * KEEP_TOPK + r) * 4;
      ob[0] = 0.f; ob[1] = 0.f; ob[2] = 0.f; ob[3] = 0.f;
      oScores[b * KEEP_TOPK + r] = 0.f;
      oLabels[b * KEEP_TOPK + r] = -1.f;
    }
  }
}

// ---------------- host launcher ----------------
extern "C" void kernel_launch(void* const* d_in, const int* in_sizes, int n_in,
                              void* d_out, int out_size, void* d_ws, size_t ws_size,
                              hipStream_t stream) {
  (void)in_sizes; (void)n_in; (void)out_size; (void)ws_size;
  const float* cls0 = (const float*)d_in[0];
  const float* bb0  = (const float*)d_in[1];
  const float* cls1 = (const float*)d_in[2];
  const float* bb1  = (const float*)d_in[3];
  const float* cls2 = (const float*)d_in[4];
  const float* bb2  = (const float*)d_in[5];
  float* out = (float*)d_out;

  // workspace carve (bytes): ~3.7 MB total
  char* ws = (char*)d_ws;
  float* pScore   = (float*)(ws + 0);        // 4*8*8400 f32 = 1,075,200 B
  int*   pLabel   = (int*)  (ws + 1075200);  // 1,075,200 B
  float* scores   = (float*)(ws + 2150400);  // 268,800 B
  int*   labels   = (int*)  (ws + 2419200);  // 268,800 B
  float* boxes    = (float*)(ws + 2688000);  // 1,075,200 B
  int*   topIdx   = (int*)  (ws + 3763200);  // 32,000 B
  float* topScore = (float*)(ws + 3795200);  // 32,000 B

  dim3 g1((NANCH + 255) / 256, B_, NCHUNK);
  k_partial_max<<<g1, 256, 0, stream>>>(cls0, cls1, cls2, pScore, pLabel);

  dim3 g2((NANCH + 255) / 256, B_);
  k_reduce_decode<<<g2, 256, 0, stream>>>(bb0, bb1, bb2, pScore, pLabel,
                                          scores, labels, boxes);

  int sortBytes = NSORT * 8;  // 128 KB (320KB LDS/WGP on CDNA5)
  hipFuncSetAttribute((const void*)k_sort, hipFuncAttributeMaxDynamicSharedMemorySize, sortBytes);
  k_sort<<<B_, 1024, sortBytes, stream>>>(scores, topIdx, topScore);

  int nmsBytes = (NPAD * 4 + NPAD + NPAD) * 4 + (NPAD * 32 + 32) * 4;  // 153,344 B
  hipFuncSetAttribute((const void*)k_nms, hipFuncAttributeMaxDynamicSharedMemorySize, nmsBytes);
  k_nms<<<B_, 1024, nmsBytes, stream>>>(topIdx, topScore, boxes, labels, out);
}